// HomogeneityLoss_27118423506994
// MI455X (gfx1250) — compile-verified
//
#include <hip/hip_runtime.h>
#include <math.h>

// ---------------------------------------------------------------------------
// HomogeneityLoss on MI455X (gfx1250, wave32)
//   center: 8192 x 64 f32.  Pairwise distances via fp32 WMMA (16x16x4),
//   fused on-chip top-17 selection per row, then log_softmax KL -> scalar.
// ---------------------------------------------------------------------------

typedef __attribute__((ext_vector_type(2))) float v2f;
typedef __attribute__((ext_vector_type(8))) float v8f;

#define NPTS 8192
#define DIM  64
#define KSEL 17              // 16 neighbors + self
#define ROWS_PER_WG 16
#define CS   512             // columns per super-block staged in LDS
#define ROWSTRIDE 560        // >= CS and >= 32*KSEL (=544); LDS row pitch
#define NWAVES 16            // 512 threads / 32
#define TILES_PER_SB (CS / 16)               // 32
#define TILES_PER_WAVE (TILES_PER_SB / NWAVES) // 2
#define NSB (NPTS / CS)      // 16

// ---------------- kernel 1: row squared norms ------------------------------
__global__ __launch_bounds__(256) void row_norms_kernel(
    const float* __restrict__ x, float* __restrict__ x2) {
  int i = blockIdx.x * blockDim.x + threadIdx.x;
  if (i < NPTS) {
    const float4* p = (const float4*)(x + (size_t)i * DIM);
    float s = 0.0f;
#pragma unroll
    for (int q = 0; q < DIM / 4; ++q) {
      float4 v = p[q];
      s += v.x * v.x + v.y * v.y + v.z * v.z + v.w * v.w;
    }
    x2[i] = s;
  }
}

// ---------------- kernel 2: WMMA distances + fused top-17 ------------------
__global__ __launch_bounds__(512, 1) void knn_mean_kernel(
    const float* __restrict__ x, const float* __restrict__ x2,
    float* __restrict__ mean_out) {
  __shared__ float lds[ROWS_PER_WG * ROWSTRIDE];

  const int tid  = threadIdx.x;
  const int lane = tid & 31;
  const int wave = tid >> 5;
  const int r0   = blockIdx.x * ROWS_PER_WG;

  // WMMA f32 16x16x4 A layout: lanes 0-15 hold M=lane with K={4c, 4c+1} in
  // vgpr pair; lanes 16-31 hold M=lane-16 with K={4c+2, 4c+3}.
  const int m16 = lane & 15;
  const int kh  = (lane >> 4) * 2;     // 0 or 2

  // Load the 16x64 f32 A strip once (held in 32 VGPRs for the whole kernel).
  v2f a[16];
#pragma unroll
  for (int c = 0; c < 16; ++c) {
    const float* p = x + (size_t)(r0 + m16) * DIM + 4 * c + kh;
    a[c] = *(const v2f*)p;
  }

  // Row norms for the epilogue: C/D layout row = r + 8*(lane>>4).
  float x2r[8];
#pragma unroll
  for (int r = 0; r < 8; ++r) x2r[r] = x2[r0 + 8 * (lane >> 4) + r];

  // Per-lane sorted (ascending) list of the KSEL smallest distances seen for
  // the row this wave owns (row index == wave).
  float list[KSEL];
#pragma unroll
  for (int s = 0; s < KSEL; ++s) list[s] = 3.0e38f;

  for (int sb = 0; sb < NSB; ++sb) {
    const int cs0 = sb * CS;

    // ---- compute phase: this wave produces TILES_PER_WAVE 16x16 tiles ----
#pragma unroll
    for (int t = 0; t < TILES_PER_WAVE; ++t) {
      const int tileLocal = wave * TILES_PER_WAVE + t;
      const int j0 = cs0 + tileLocal * 16;

      // B tile (64x16 = columns' rows, same K striping as A).
      v2f b[16];
#pragma unroll
      for (int c = 0; c < 16; ++c) {
        const float* p = x + (size_t)(j0 + m16) * DIM + 4 * c + kh;
        b[c] = *(const v2f*)p;
      }

      v8f acc = {};
#pragma unroll
      for (int c = 0; c < 16; ++c) {
        acc = __builtin_amdgcn_wmma_f32_16x16x4_f32(
            /*neg_a=*/false, a[c], /*neg_b=*/false, b[c],
            /*c_mod=*/(short)0, acc, /*reuse_a=*/false, /*reuse_b=*/false);
      }

      const float x2c     = x2[j0 + m16];
      const int colLocal  = tileLocal * 16 + m16;
#pragma unroll
      for (int r = 0; r < 8; ++r) {
        float d2 = x2r[r] + x2c - 2.0f * acc[r];
        float dd = __builtin_sqrtf(fmaxf(d2, 0.0f));
        int row  = r + 8 * (lane >> 4);
        lds[row * ROWSTRIDE + colLocal] = dd;
      }
    }
    __syncthreads();

    // ---- selection phase: wave w scans row w's CS staged distances ----
#pragma unroll
    for (int i = 0; i < CS / 32; ++i) {
      float v = lds[wave * ROWSTRIDE + lane + 32 * i];
      if (v < list[KSEL - 1]) {
        list[KSEL - 1] = v;
#pragma unroll
        for (int s = KSEL - 1; s > 0; --s) {   // bubble down, stays sorted
          float lo = list[s - 1], hi = list[s];
          list[s - 1] = fminf(lo, hi);
          list[s]     = fmaxf(lo, hi);
        }
      }
    }
    __syncthreads();
  }

  // ---- merge: 32 per-lane lists -> row top-17 -> mean of 16 neighbors ----
#pragma unroll
  for (int s = 0; s < KSEL; ++s)
    lds[wave * ROWSTRIDE + lane * KSEL + s] = list[s];
  __syncthreads();

  if (lane == 0) {
    float fin[KSEL];
#pragma unroll
    for (int s = 0; s < KSEL; ++s) fin[s] = 3.0e38f;
    for (int q = 0; q < 32 * KSEL; ++q) {
      float v = lds[wave * ROWSTRIDE + q];
      if (v < fin[KSEL - 1]) {
        fin[KSEL - 1] = v;
#pragma unroll
        for (int s = KSEL - 1; s > 0; --s) {
          float lo = fin[s - 1], hi = fin[s];
          fin[s - 1] = fminf(lo, hi);
          fin[s]     = fmaxf(lo, hi);
        }
      }
    }
    float sum = 0.0f;
#pragma unroll
    for (int s = 0; s < KSEL; ++s) sum += fin[s];
    // drop the smallest (self, d ~ 0), average the 16 true neighbors
    mean_out[r0 + wave] = (sum - fin[0]) * (1.0f / 16.0f);
  }
}

// ---------------- kernel 3: log_softmax KL vs uniform -> scalar ------------
__global__ __launch_bounds__(1024) void kl_loss_kernel(
    const float* __restrict__ md, float* __restrict__ out) {
  __shared__ float red[1024];
  const int tid = threadIdx.x;

  float mx = -3.0e38f;
  for (int i = tid; i < NPTS; i += 1024) mx = fmaxf(mx, md[i]);
  red[tid] = mx;
  __syncthreads();
  for (int s = 512; s > 0; s >>= 1) {
    if (tid < s) red[tid] = fmaxf(red[tid], red[tid + s]);
    __syncthreads();
  }
  mx = red[0];
  __syncthreads();

  float se = 0.0f, sm = 0.0f;
  for (int i = tid; i < NPTS; i += 1024) {
    float v = md[i];
    se += expf(v - mx);
    sm += v;
  }
  red[tid] = se;
  __syncthreads();
  for (int s = 512; s > 0; s >>= 1) {
    if (tid < s) red[tid] += red[tid + s];
    __syncthreads();
  }
  se = red[0];
  __syncthreads();
  red[tid] = sm;
  __syncthreads();
  for (int s = 512; s > 0; s >>= 1) {
    if (tid < s) red[tid] += red[tid + s];
    __syncthreads();
  }
  if (tid == 0) {
    float sumM = red[0];
    float lse  = mx + logf(se);
    // loss = -ln(n) - mean(m) + logsumexp(m)
    out[0] = -logf((float)NPTS) - sumM / (float)NPTS + lse;
  }
}

// ---------------------------------------------------------------------------
extern "C" void kernel_launch(void* const* d_in, const int* in_sizes, int n_in,
                              void* d_out, int out_size, void* d_ws,
                              size_t ws_size, hipStream_t stream) {
  (void)in_sizes; (void)n_in; (void)out_size; (void)ws_size;

  const float* center = (const float*)d_in[0];
  float* out = (float*)d_out;

  float* x2 = (float*)d_ws;          // 8192 floats
  float* md = x2 + NPTS;             // 8192 floats (mean_dists)

  row_norms_kernel<<<NPTS / 256, 256, 0, stream>>>(center, x2);
  knn_mean_kernel<<<NPTS / ROWS_PER_WG, 512, 0, stream>>>(center, x2, md);
  kl_loss_kernel<<<1, 1024, 0, stream>>>(md, out);
}